// AttentionForONNX_87479893885149
// MI455X (gfx1250) — compile-verified
//
#include <hip/hip_runtime.h>
#include <hip/hip_bf16.h>
#include <cstddef>

typedef float v8f __attribute__((ext_vector_type(8)));
typedef float v2f __attribute__((ext_vector_type(2)));

#define T_Q 4
#define B_B 32
#define E_E 1024
#define H_H 16
#define HD 64
#define CACHE 4096
#define S_S 4100   // CACHE + T
#define SCALING 0.125f  // 1/sqrt(64)

// ---------------------------------------------------------------------------
// Kernel 1: QKV projection.  x:[128,1024] (rows = t*32+b), W row-major [out,in]
// computes (x @ W^T + b) (* scaling for q) and scatters into [B*H, T, HD].
// One wave per 16x16 output tile; K swept in steps of 4 via V_WMMA_F32_16X16X4_F32.
// ---------------------------------------------------------------------------
__global__ __launch_bounds__(32) void qkv_proj_kernel(
    const float* __restrict__ x,
    const float* __restrict__ Wq, const float* __restrict__ bq,
    const float* __restrict__ Wk, const float* __restrict__ bk,
    const float* __restrict__ Wv, const float* __restrict__ bv,
    float* __restrict__ q_s, float* __restrict__ k_s, float* __restrict__ v_s)
{
    const int mt = blockIdx.x;          // 0..7   (M tiles over 128 rows)
    const int nt = blockIdx.y;          // 0..63  (N tiles over 1024 feats)
    const int which = blockIdx.z;       // 0=q 1=k 2=v

    const float* W  = (which == 0) ? Wq : (which == 1) ? Wk : Wv;
    const float* bb = (which == 0) ? bq : (which == 1) ? bk : bv;
    float* dst      = (which == 0) ? q_s : (which == 1) ? k_s : v_s;

    const int lane  = threadIdx.x;
    const int n16   = lane & 15;
    const int half2 = (lane >> 4) * 2;  // K sub-offset 0 or 2

    const float* arow = x + (size_t)(mt * 16 + n16) * E_E;   // A row (M = lane%16)
    const float* brow = W + (size_t)(nt * 16 + n16) * E_E;   // B col n = W row n

    v8f acc = {};
    for (int kk = 0; kk < E_E / 4; ++kk) {
        const int c = 4 * kk + half2;
        v2f a = *(const v2f*)(arow + c);
        v2f b = *(const v2f*)(brow + c);
        acc = __builtin_amdgcn_wmma_f32_16x16x4_f32(
            false, a, false, b, (short)0, acc, false, false);
    }

    const int n = nt * 16 + n16;
    const float bias = bb[n];
    const float sc = (which == 0) ? SCALING : 1.0f;
    const int h = n >> 6, d = n & 63;
#pragma unroll
    for (int r = 0; r < 8; ++r) {
        const int m = mt * 16 + r + 8 * (lane >> 4);  // D row
        const float val = (acc[r] + bias) * sc;
        const int t = m >> 5, b_ = m & 31;
        dst[(((size_t)(b_ * H_H + h)) * T_Q + t) * HD + d] = val;
    }
}

// ---------------------------------------------------------------------------
// Kernel 2: flash-style attention per (b,h).  8 waves, each owns a contiguous
// range of 16-key tiles; online softmax in registers; P@V via WMMA; wave
// partials merged through LDS.  Streams K/V cache exactly once.
// ---------------------------------------------------------------------------
__global__ __launch_bounds__(256) void attn_kernel(
    const float* __restrict__ q_s, const float* __restrict__ k_s,
    const float* __restrict__ v_s,
    const float* __restrict__ k_cache, const float* __restrict__ v_cache,
    const unsigned char* __restrict__ mask,   // [B, S] bool
    float* __restrict__ o_s)
{
    const int bh   = blockIdx.x;       // 0..511
    const int b    = bh >> 4;          // / H
    const int lane = threadIdx.x & 31;
    const int w    = threadIdx.x >> 5; // wave 0..7
    const int half = lane >> 4;
    const int n16  = lane & 15;

    __shared__ float sP[8][16][17];    // per-wave P-tile transpose patch
    __shared__ float sm[8][4];
    __shared__ float sl[8][4];
    __shared__ float so[8][4][64];

    // ---- preload q A-fragments (reused for every key tile) ----
    v2f aq[16];
    {
        const float* qb = q_s + (size_t)bh * T_Q * HD;
        const int t = n16;  // A-matrix row M = lane%16 (rows >= T are zero pad)
#pragma unroll
        for (int kk = 0; kk < 16; ++kk) {
            const int c = 4 * kk + 2 * half;
            v2f a = {};
            if (t < T_Q) a = *(const v2f*)(qb + t * HD + c);
            aq[kk] = a;
        }
    }

    float row_m[8], row_l[8];
    v8f od[4];
#pragma unroll
    for (int r = 0; r < 8; ++r) { row_m[r] = -1e30f; row_l[r] = 0.f; }
#pragma unroll
    for (int t = 0; t < 4; ++t) od[t] = (v8f){};

    const float* kbase_c = k_cache + (size_t)bh * CACHE * HD;
    const float* vbase_c = v_cache + (size_t)bh * CACHE * HD;
    const float* kbase_n = k_s + (size_t)bh * T_Q * HD;
    const float* vbase_n = v_s + (size_t)bh * T_Q * HD;
    const unsigned char* mrow = mask + (size_t)b * S_S;

    const int NTILES = (S_S + 15) / 16;          // 257
    const int t0 = (NTILES * w) / 8;
    const int t1 = (NTILES * (w + 1)) / 8;

    for (int tile = t0; tile < t1; ++tile) {
        const int j0 = tile * 16;
        const int j  = j0 + n16;                 // key index for this lane (B col)
        const int jc = (j < S_S) ? j : (S_S - 1);
        const float* krow = (jc < CACHE) ? (kbase_c + (size_t)jc * HD)
                                         : (kbase_n + (size_t)(jc - CACHE) * HD);
        const bool colmask = (j >= S_S) || (mrow[jc] != 0);

        // prefetch next tile's K row for this lane (speculative, may be OOB)
        __builtin_prefetch(krow + 16 * HD, 0, 0);

        // ---- scores: 16 x WMMA f32 16x16x4 over HD=64 ----
        v8f acc = {};
#pragma unroll
        for (int kk = 0; kk < 16; ++kk) {
            const int c = 4 * kk + 2 * half;
            v2f bf = *(const v2f*)(krow + c);
            acc = __builtin_amdgcn_wmma_f32_16x16x4_f32(
                false, aq[kk], false, bf, (short)0, acc, false, false);
        }

        // ---- online softmax update (per D row r; lane-half reductions) ----
#pragma unroll
        for (int r = 0; r < 8; ++r) {
            float s = colmask ? -1e30f : acc[r];
            float tm = s;
            tm = fmaxf(tm, __shfl_xor(tm, 1, 16));
            tm = fmaxf(tm, __shfl_xor(tm, 2, 16));
            tm = fmaxf(tm, __shfl_xor(tm, 4, 16));
            tm = fmaxf(tm, __shfl_xor(tm, 8, 16));
            const float mo = row_m[r];
            const float mn = fmaxf(mo, tm);
            const float scale = __expf(mo - mn);
            const float p = colmask ? 0.f : __expf(s - mn);
            float ps = p;
            ps += __shfl_xor(ps, 1, 16);
            ps += __shfl_xor(ps, 2, 16);
            ps += __shfl_xor(ps, 4, 16);
            ps += __shfl_xor(ps, 8, 16);
            row_l[r] = row_l[r] * scale + ps;
            row_m[r] = mn;
            od[0][r] *= scale; od[1][r] *= scale;
            od[2][r] *= scale; od[3][r] *= scale;
            sP[w][r + 8 * half][n16] = p;   // stage P for transpose to A-layout
        }
        asm volatile("s_wait_dscnt 0" ::: "memory");  // per-wave LDS ordering

        // ---- P @ V: K-dim = 16 keys -> 4 WMMA steps, 4 N-tiles of HD ----
#pragma unroll
        for (int kk = 0; kk < 4; ++kk) {
            const int c = 4 * kk + 2 * half;
            v2f ap = { sP[w][n16][c], sP[w][n16][c + 1] };
            const int j1 = j0 + c, j2 = j1 + 1;
            const int j1c = (j1 < S_S) ? j1 : (S_S - 1);
            const int j2c = (j2 < S_S) ? j2 : (S_S - 1);
            const float* vr1 = (j1c < CACHE) ? (vbase_c + (size_t)j1c * HD)
                                             : (vbase_n + (size_t)(j1c - CACHE) * HD);
            const float* vr2 = (j2c < CACHE) ? (vbase_c + (size_t)j2c * HD)
                                             : (vbase_n + (size_t)(j2c - CACHE) * HD);
#pragma unroll
            for (int t = 0; t < 4; ++t) {
                v2f bv2 = { vr1[t * 16 + n16], vr2[t * 16 + n16] };
                od[t] = __builtin_amdgcn_wmma_f32_16x16x4_f32(
                    false, ap, false, bv2, (short)0, od[t], false, false);
            }
        }
    }

    // ---- publish wave partials (rows 0..3 live in half 0, VGPRs 0..3) ----
    if (lane < 16) {
#pragma unroll
        for (int r = 0; r < 4; ++r) {
            if (lane == 0) { sm[w][r] = row_m[r]; sl[w][r] = row_l[r]; }
#pragma unroll
            for (int t = 0; t < 4; ++t) so[w][r][t * 16 + lane] = od[t][r];
        }
    }
    __syncthreads();

    // ---- merge 8 wave partials; 256 threads -> 4 rows x 64 dims ----
    const int tid = threadIdx.x;
    const int row = tid >> 6, d = tid & 63;
    float M = -1e30f;
#pragma unroll
    for (int ww = 0; ww < 8; ++ww) M = fmaxf(M, sm[ww][row]);
    float L = 0.f, O = 0.f;
#pragma unroll
    for (int ww = 0; ww < 8; ++ww) {
        const float e = __expf(sm[ww][row] - M);
        L += sl[ww][row] * e;
        O += so[ww][row][d] * e;
    }
    o_s[((size_t)bh * T_Q + row) * HD + d] = (L > 0.f) ? (O / L) : 0.f;
}

// ---------------------------------------------------------------------------
// Kernel 3: output projection.  x2[m=t*32+b][e] = o_s[b*H + e/64][t][e%64];
// out = x2 @ Wo^T + bo, written as [T,B,E] (== row m, col n).
// ---------------------------------------------------------------------------
__global__ __launch_bounds__(32) void out_proj_kernel(
    const float* __restrict__ o_s,
    const float* __restrict__ Wo, const float* __restrict__ bo,
    float* __restrict__ out)
{
    const int mt = blockIdx.x;     // 0..7
    const int nt = blockIdx.y;     // 0..63
    const int lane  = threadIdx.x;
    const int n16   = lane & 15;
    const int half2 = (lane >> 4) * 2;

    const int m  = mt * 16 + n16;          // A row
    const int tq = m >> 5, b = m & 31;
    const int nrow = nt * 16 + n16;        // B "column" = Wo row
    const float* brow = Wo + (size_t)nrow * E_E;
    const float* obase = o_s + (size_t)b * H_H * T_Q * HD + (size_t)tq * HD;

    v8f acc = {};
    for (int kk = 0; kk < E_E / 4; ++kk) {
        const int c = 4 * kk + half2;      // even, c and c+1 share a head
        const float* arow = obase + (size_t)(c >> 6) * (T_Q * HD) + (c & 63);
        v2f a = *(const v2f*)arow;
        v2f bb = *(const v2f*)(brow + c);
        acc = __builtin_amdgcn_wmma_f32_16x16x4_f32(
            false, a, false, bb, (short)0, acc, false, false);
    }

    const float bias = bo[nrow];
#pragma unroll
    for (int r = 0; r < 8; ++r) {
        const int m2 = mt * 16 + r + 8 * (lane >> 4);
        out[(size_t)m2 * E_E + nrow] = acc[r] + bias;
    }
}

// ---------------------------------------------------------------------------
extern "C" void kernel_launch(void* const* d_in, const int* in_sizes, int n_in,
                              void* d_out, int out_size, void* d_ws, size_t ws_size,
                              hipStream_t stream)
{
    (void)in_sizes; (void)n_in; (void)out_size; (void)ws_size;
    const float* query = (const float*)d_in[0];
    /* d_in[1] ("key") is unused by the reference decode path */
    const unsigned char* mask = (const unsigned char*)d_in[2];  // jax bool = 1 byte
    const float* k_cache = (const float*)d_in[3];
    const float* v_cache = (const float*)d_in[4];
    const float* Wq = (const float*)d_in[5];  const float* bq = (const float*)d_in[6];
    const float* Wk = (const float*)d_in[7];  const float* bk = (const float*)d_in[8];
    const float* Wv = (const float*)d_in[9];  const float* bv = (const float*)d_in[10];
    const float* Wo = (const float*)d_in[11]; const float* bo = (const float*)d_in[12];
    float* out = (float*)d_out;

    char* ws = (char*)d_ws;
    float* q_s = (float*)(ws + 0);              // 512 KB each: [B*H, T, HD]
    float* k_s = (float*)(ws + 512 * 1024);
    float* v_s = (float*)(ws + 1024 * 1024);
    float* o_s = (float*)(ws + 1536 * 1024);

    qkv_proj_kernel<<<dim3(8, 64, 3), 32, 0, stream>>>(
        query, Wq, bq, Wk, bk, Wv, bv, q_s, k_s, v_s);

    attn_kernel<<<dim3(B_B * H_H), 256, 0, stream>>>(
        q_s, k_s, v_s, k_cache, v_cache, mask, o_s);

    out_proj_kernel<<<dim3(8, 64), 32, 0, stream>>>(o_s, Wo, bo, out);
}